// GNNModel_23192823399174
// MI455X (gfx1250) — compile-verified
//
#include <hip/hip_runtime.h>
#include <hip/hip_bf16.h>

// 2-layer GCN: out = GCN2(relu(GCN1(x))), symmetric-normalized with self-loops.
// N = 100000 nodes, E = 1.6M edges, F: 6 -> 64 -> 8, all fp32.
// Edge index is int64 per the reference (jnp.int64).

typedef float v2f __attribute__((ext_vector_type(2)));
typedef float v8f __attribute__((ext_vector_type(8)));

#define F1 64
#define F2 8

// ---------------- degree / normalization ----------------

__global__ void k_deg_init(float* __restrict__ deg, int n) {
    int i = blockIdx.x * blockDim.x + threadIdx.x;
    if (i < n) deg[i] = 1.0f;  // +1 self-loop
}

__global__ void k_deg_count(const long long* __restrict__ dst, float* __restrict__ deg, int e) {
    int i = blockIdx.x * blockDim.x + threadIdx.x;
    if (i < e) atomicAdd(&deg[(int)dst[i]], 1.0f);
}

__global__ void k_rsqrt_inplace(float* __restrict__ deg, int n) {
    int i = blockIdx.x * blockDim.x + threadIdx.x;
    if (i < n) deg[i] = rsqrtf(deg[i]);
}

// ---------------- GEMM1: h1[N,64] = x[N,6] @ W1[6,64] via V_WMMA_F32_16X16X4_F32 ----
//
// One wave handles 16 nodes x 64 output cols. K=6 split into two K=4 steps
// (second step zero-padded). A layout (32-bit 16x4): lanes 0-15 hold K=0,1 in
// the 2 VGPRs; lanes 16-31 hold K=2,3. B layout (4x16): VGPR0 = row (hi?2:0),
// VGPR1 = row (hi?3:1), col = lane%16. C/D: VGPR v, lane L -> M = v + (L<16?0:8).
//
// All loads use unconditional in-bounds addresses; lane-dependent zeroing is a
// data select (v_cndmask), never an EXEC-predicated load.
__global__ void k_gemm1_wmma(const float* __restrict__ x, const float* __restrict__ W1,
                             float* __restrict__ h1, int ntiles) {
    int wave = (blockIdx.x * blockDim.x + threadIdx.x) >> 5;
    int lane = threadIdx.x & 31;
    if (wave >= ntiles) return;  // wave-uniform exit; EXEC all-1s for survivors
    const int row0 = wave * 16;
    const bool hi  = lane >= 16;
    const int  col = lane & 15;
    const int  m   = row0 + (lane & 15);
    const int  k0  = hi ? 2 : 0;  // K rows held by this half-wave in step 0

    const float* xr = x + (size_t)m * 6;
    // A, K-step 0: per-lane address select (all in-bounds).
    v2f a0;
    a0.x = xr[k0];
    a0.y = xr[k0 + 1];
    // A, K-step 1 (K=4..7): rows 6,7 don't exist -> hi lanes carry zeros.
    float x4 = xr[4], x5 = xr[5];
    v2f a1;
    a1.x = hi ? 0.0f : x4;
    a1.y = hi ? 0.0f : x5;

#pragma unroll
    for (int t = 0; t < 4; ++t) {
        const int n0 = t * 16;
        v2f b0, b1;
        b0.x = W1[(k0 + 0) * F1 + n0 + col];
        b0.y = W1[(k0 + 1) * F1 + n0 + col];
        float w4 = W1[4 * F1 + n0 + col];
        float w5 = W1[5 * F1 + n0 + col];
        b1.x = hi ? 0.0f : w4;
        b1.y = hi ? 0.0f : w5;
        v8f c = {};
        c = __builtin_amdgcn_wmma_f32_16x16x4_f32(false, a0, false, b0, (short)0, c, false, false);
        c = __builtin_amdgcn_wmma_f32_16x16x4_f32(false, a1, false, b1, (short)0, c, false, false);
        const int rbase = row0 + (hi ? 8 : 0);
#pragma unroll
        for (int v = 0; v < 8; ++v)
            h1[(size_t)(rbase + v) * F1 + n0 + col] = c[v];
    }
}

// Scalar tail (only launched if N % 16 != 0; N=100000 is divisible, so unused).
__global__ void k_gemm1_tail(const float* __restrict__ x, const float* __restrict__ W1,
                             float* __restrict__ h1, int n0, int n) {
    int gid = blockIdx.x * blockDim.x + threadIdx.x;
    int i = n0 + (gid >> 6);
    int f = gid & 63;
    if (i >= n) return;
    float s = 0.f;
#pragma unroll
    for (int k = 0; k < 6; ++k) s += x[(size_t)i * 6 + k] * W1[k * F1 + f];
    h1[(size_t)i * F1 + f] = s;
}

// ---------------- layer-1 aggregation ----------------

// agg1 = b1 + h1 * dinv^2 (self-loop term), then edge atomics add on top.
__global__ void k_init_agg1(const float* __restrict__ h1, const float* __restrict__ dinv,
                            const float* __restrict__ b1, float* __restrict__ agg1, int n) {
    int gid = blockIdx.x * blockDim.x + threadIdx.x;
    if (gid >= n * F1) return;
    int i = gid >> 6, f = gid & 63;
    float di = dinv[i];
    agg1[gid] = b1[f] + h1[gid] * di * di;
}

// 16 threads per edge, float4 gather + 4 non-returning f32 atomics each.
// The 16 threads sharing an edge issue identical index/dinv loads, which
// coalesce into broadcast requests and hit L2 (working set < 192 MB L2).
__global__ void k_edge1(const long long* __restrict__ src, const long long* __restrict__ dst,
                        const float* __restrict__ dinv, const float* __restrict__ h1,
                        float* __restrict__ agg1, int e) {
    int gid = blockIdx.x * blockDim.x + threadIdx.x;
    int eid = gid >> 4;
    if (eid >= e) return;
    int f0 = (gid & 15) << 2;
    int s = (int)src[eid], d = (int)dst[eid];
    float coef = dinv[s] * dinv[d];
    const float4 hv = *(const float4*)(h1 + (size_t)s * F1 + f0);
    float* out = agg1 + (size_t)d * F1 + f0;
    atomicAdd(out + 0, hv.x * coef);
    atomicAdd(out + 1, hv.y * coef);
    atomicAdd(out + 2, hv.z * coef);
    atomicAdd(out + 3, hv.w * coef);
}

__global__ void k_relu(float* __restrict__ a, int cnt) {
    int gid = blockIdx.x * blockDim.x + threadIdx.x;
    if (gid < cnt) a[gid] = fmaxf(a[gid], 0.0f);
}

// ---------------- GEMM2: h2[N,8] = hrelu[N,64] @ W2[64,8] via WMMA ----------------
//
// K=64 -> 16 K-steps of 4; B cols 8..15 are zero-padding: the address is
// clamped to col 7 (in-bounds) and the data zero-selected, so no EXEC-masked
// loads. Only cols 0..7 are stored.
__global__ void k_gemm2_wmma(const float* __restrict__ h, const float* __restrict__ W2,
                             float* __restrict__ h2, int ntiles) {
    int wave = (blockIdx.x * blockDim.x + threadIdx.x) >> 5;
    int lane = threadIdx.x & 31;
    if (wave >= ntiles) return;
    const int row0 = wave * 16;
    const bool hi  = lane >= 16;
    const int  col = lane & 15;
    const int  m   = row0 + (lane & 15);
    const float* hr = h + (size_t)m * F1;
    const bool valid_col = col < F2;
    const int  colc = valid_col ? col : (F2 - 1);  // clamped, always in-bounds

    v8f c = {};
#pragma unroll
    for (int k = 0; k < 16; ++k) {
        const int ka = 4 * k + (hi ? 2 : 0);
        v2f a, b;
        a.x = hr[ka];
        a.y = hr[ka + 1];
        float w0 = W2[ka * F2 + colc];
        float w1 = W2[(ka + 1) * F2 + colc];
        b.x = valid_col ? w0 : 0.0f;
        b.y = valid_col ? w1 : 0.0f;
        c = __builtin_amdgcn_wmma_f32_16x16x4_f32(false, a, false, b, (short)0, c, false, false);
    }
    if (valid_col) {
        const int rbase = row0 + (hi ? 8 : 0);
#pragma unroll
        for (int v = 0; v < 8; ++v)
            h2[(size_t)(rbase + v) * F2 + col] = c[v];
    }
}

__global__ void k_gemm2_tail(const float* __restrict__ h, const float* __restrict__ W2,
                             float* __restrict__ h2, int n0, int n) {
    int gid = blockIdx.x * blockDim.x + threadIdx.x;
    int i = n0 + (gid >> 3);
    int f = gid & 7;
    if (i >= n) return;
    float s = 0.f;
    for (int k = 0; k < F1; ++k) s += h[(size_t)i * F1 + k] * W2[k * F2 + f];
    h2[(size_t)i * F2 + f] = s;
}

// ---------------- layer-2 aggregation ----------------

__global__ void k_init_out(const float* __restrict__ h2, const float* __restrict__ dinv,
                           const float* __restrict__ b2, float* __restrict__ out, int n) {
    int gid = blockIdx.x * blockDim.x + threadIdx.x;
    if (gid >= n * F2) return;
    int i = gid >> 3, f = gid & 7;
    float di = dinv[i];
    out[gid] = b2[f] + h2[gid] * di * di;
}

// 2 threads per edge, float4 gather + 4 atomics each (8 features).
__global__ void k_edge2(const long long* __restrict__ src, const long long* __restrict__ dst,
                        const float* __restrict__ dinv, const float* __restrict__ h2,
                        float* __restrict__ out, int e) {
    int gid = blockIdx.x * blockDim.x + threadIdx.x;
    int eid = gid >> 1;
    if (eid >= e) return;
    int f0 = (gid & 1) << 2;
    int s = (int)src[eid], d = (int)dst[eid];
    float coef = dinv[s] * dinv[d];
    const float4 hv = *(const float4*)(h2 + (size_t)s * F2 + f0);
    float* o = out + (size_t)d * F2 + f0;
    atomicAdd(o + 0, hv.x * coef);
    atomicAdd(o + 1, hv.y * coef);
    atomicAdd(o + 2, hv.z * coef);
    atomicAdd(o + 3, hv.w * coef);
}

// ---------------- launch ----------------

extern "C" void kernel_launch(void* const* d_in, const int* in_sizes, int n_in,
                              void* d_out, int out_size, void* d_ws, size_t ws_size,
                              hipStream_t stream) {
    const float*     x  = (const float*)d_in[0];
    const long long* ei = (const long long*)d_in[1];   // int64 per reference
    const float*     W1 = (const float*)d_in[2];
    const float*     b1 = (const float*)d_in[3];
    const float*     W2 = (const float*)d_in[4];
    const float*     b2 = (const float*)d_in[5];
    float*           out = (float*)d_out;

    const int n = in_sizes[0] / 6;       // 100000
    const int e = in_sizes[1] / 2;       // 1600000
    const long long* src = ei;
    const long long* dst = ei + e;

    // Workspace layout (floats). All offsets 16B-aligned (n % 4 == 0).
    float* ws   = (float*)d_ws;
    float* dinv = ws;                     // [n]
    float* h1   = dinv + n;               // [n*64]
    float* agg1 = h1 + (size_t)n * F1;    // [n*64]
    float* h2   = agg1 + (size_t)n * F1;  // [n*8]

    const int B = 256;
    const int ntiles = n / 16;
    const int tail   = n - ntiles * 16;

    // 1) degrees with self-loop, then dinv = rsqrt(deg)
    k_deg_init<<<(n + B - 1) / B, B, 0, stream>>>(dinv, n);
    k_deg_count<<<(e + B - 1) / B, B, 0, stream>>>(dst, dinv, e);
    k_rsqrt_inplace<<<(n + B - 1) / B, B, 0, stream>>>(dinv, n);

    // 2) h1 = x @ W1 (WMMA f32 16x16x4)
    if (ntiles > 0) {
        int waves_per_block = B / 32;
        int blocks = (ntiles + waves_per_block - 1) / waves_per_block;
        k_gemm1_wmma<<<blocks, B, 0, stream>>>(x, W1, h1, ntiles);
    }
    if (tail > 0)
        k_gemm1_tail<<<(tail * F1 + B - 1) / B, B, 0, stream>>>(x, W1, h1, ntiles * 16, n);

    // 3) agg1 = b1 + self-loop term; edge scatter-add; ReLU
    k_init_agg1<<<((long long)n * F1 + B - 1) / B, B, 0, stream>>>(h1, dinv, b1, agg1, n);
    k_edge1<<<((long long)e * 16 + B - 1) / B, B, 0, stream>>>(src, dst, dinv, h1, agg1, e);
    k_relu<<<((long long)n * F1 + B - 1) / B, B, 0, stream>>>(agg1, n * F1);

    // 4) h2 = relu(agg1) @ W2 (WMMA f32 16x16x4, K=64)
    if (ntiles > 0) {
        int waves_per_block = B / 32;
        int blocks = (ntiles + waves_per_block - 1) / waves_per_block;
        k_gemm2_wmma<<<blocks, B, 0, stream>>>(agg1, W2, h2, ntiles);
    }
    if (tail > 0)
        k_gemm2_tail<<<(tail * F2 + B - 1) / B, B, 0, stream>>>(agg1, W2, h2, ntiles * 16, n);

    // 5) out = b2 + self-loop term; edge scatter-add into d_out
    k_init_out<<<((long long)n * F2 + B - 1) / B, B, 0, stream>>>(h2, dinv, b2, out, n);
    k_edge2<<<((long long)e * 2 + B - 1) / B, B, 0, stream>>>(src, dst, dinv, h2, out, e);
}